// PamDelayModel_70703751627131
// MI455X (gfx1250) — compile-verified
//
#include <hip/hip_runtime.h>
#include <cstdint>

#define DT_F       0.001f
#define BUF_LEN    201
#define NP         9
#define CHANGE_EPS 1e-4f

#define TPB  256   // 8 wave32 per block
#define EPT  4     // channels per thread (float4 streaming loads)
#define LUTN (NP + 2 * NP * NP)   // 9 + 81 + 81 = 171 floats

__device__ __forceinline__ int mod_buf(int x) {
    int r = x % BUF_LEN;
    return (r < 0) ? r + BUF_LEN : r;
}

__global__ __launch_bounds__(TPB) void pam_delay_kernel(
    const float* __restrict__ target,
    const float* __restrict__ current,
    const float* __restrict__ prev_t,
    const float* __restrict__ latch_in,
    const float* __restrict__ p_axis,
    const float* __restrict__ tau_table,
    const float* __restrict__ dead_table,
    const float* __restrict__ delay_buf,
    const int*   __restrict__ write_idx_p,
    float* __restrict__ out,
    int n)
{
    __shared__ float s_lut[LUTN];   // [0..8]=axis, [9..89]=tau, [90..170]=dead
    const int tid = threadIdx.x;

    // ---- Stage LUTs into LDS with CDNA5 async global->LDS copies (ASYNCcnt) ----
    if (tid < LUTN) {
        const float* src;
        if (tid < NP)                src = p_axis    + tid;
        else if (tid < NP + NP * NP) src = tau_table + (tid - NP);
        else                         src = dead_table + (tid - NP - NP * NP);
        uint32_t laddr = (uint32_t)(uintptr_t)(&s_lut[tid]);   // low 32 bits = LDS offset
        uint64_t gaddr = (uint64_t)(uintptr_t)src;
        asm volatile("global_load_async_to_lds_b32 %0, %1, off"
                     :: "v"(laddr), "v"(gaddr) : "memory");
    }

    // Overlap: scalar write_idx load + address setup while the async copy flies.
    const int widx = write_idx_p[0];

#if __has_builtin(__builtin_amdgcn_s_wait_asynccnt)
    __builtin_amdgcn_s_wait_asynccnt(0);
#else
    asm volatile("s_wait_asynccnt 0x0" ::: "memory");
#endif
    __syncthreads();

    const float* s_ax   = s_lut;
    const float* s_tau  = s_lut + NP;
    const float* s_dead = s_lut + NP + NP * NP;
    const float  a_lo   = s_ax[0];
    const float  a_hi   = s_ax[NP - 1];

    const int base = (blockIdx.x * TPB + tid) * EPT;
    if (base >= n) return;

    float tg[EPT], cu[EPT], pv[EPT], lt[EPT], rs[EPT];
    const bool full = (base + EPT) <= n;
    if (full) {
        *(float4*)tg = *(const float4*)(target   + base);
        *(float4*)cu = *(const float4*)(current  + base);
        *(float4*)pv = *(const float4*)(prev_t   + base);
        *(float4*)lt = *(const float4*)(latch_in + base);
    } else {
        for (int j = 0; j < EPT; ++j) {
            int i = base + j < n ? base + j : n - 1;
            tg[j] = target[i]; cu[j] = current[i]; pv[j] = prev_t[i]; lt[j] = latch_in[i];
        }
    }

#pragma unroll
    for (int j = 0; j < EPT; ++j) {
        const float tgt = tg[j];
        const float cur = cu[j];

        // 1) masked latch of start pressure on target change
        const float latch = (fabsf(tgt - pv[j]) > CHANGE_EPS) ? cur : lt[j];

        // 2) bilinear LUT lookup (shared ix/iy/weights for both tables)
        float xq = fminf(fmaxf(tgt,   a_lo), a_hi);
        float yq = fminf(fmaxf(latch, a_lo), a_hi);
        int ix = -1, iy = -1;
#pragma unroll
        for (int k = 0; k < NP; ++k) {        // searchsorted(side='right') - 1
            ix += (xq >= s_ax[k]) ? 1 : 0;
            iy += (yq >= s_ax[k]) ? 1 : 0;
        }
        ix = (ix > NP - 2) ? NP - 2 : ix;     // clamp high (low clamp implied by xq>=a_lo)
        iy = (iy > NP - 2) ? NP - 2 : iy;
        const float x0 = s_ax[ix], x1 = s_ax[ix + 1];
        const float y0 = s_ax[iy], y1 = s_ax[iy + 1];
        const float tx = (xq - x0) / (x1 - x0);
        const float ty = (yq - y0) / (y1 - y0);
        const float w00 = (1.0f - tx) * (1.0f - ty);
        const float w10 = tx * (1.0f - ty);
        const float w01 = (1.0f - tx) * ty;
        const float w11 = tx * ty;
        const int b = ix * NP + iy;           // f10 = [ix+1,iy] -> b+NP ; f01 = [ix,iy+1] -> b+1
        const float tau = w00 * s_tau[b]  + w10 * s_tau[b + NP]
                        + w01 * s_tau[b + 1] + w11 * s_tau[b + NP + 1];
        const float L   = w00 * s_dead[b] + w10 * s_dead[b + NP]
                        + w01 * s_dead[b + 1] + w11 * s_dead[b + NP + 1];

        // 3) fractional-delay read from ring buffer (slot widx logically holds tgt)
        float d = L / DT_F;
        d = fminf(fmaxf(d, 0.0f), (float)(BUF_LEN - 1));
        const int   i0   = (int)floorf(d);
        const float frac = d - (float)i0;
        const int p0 = mod_buf(widx - i0);
        const int p1 = mod_buf(widx - i0 - 1);
        const float* row = delay_buf + (size_t)(base + j) * BUF_LEN;
        const float v0 = (p0 == widx) ? tgt : row[p0];
        const float v1 = (p1 == widx) ? tgt : row[p1];
        const float p_delayed = (1.0f - frac) * v0 + frac * v1;

        // 4) first-order lag
        const float alpha = 1.0f - expf(-DT_F / tau);
        rs[j] = cur + alpha * (p_delayed - cur);
    }

    if (full) {
        *(float4*)(out + base) = *(const float4*)rs;
    } else {
        for (int j = 0; j < EPT && base + j < n; ++j) out[base + j] = rs[j];
    }
}

extern "C" void kernel_launch(void* const* d_in, const int* in_sizes, int n_in,
                              void* d_out, int out_size, void* d_ws, size_t ws_size,
                              hipStream_t stream) {
    const float* target   = (const float*)d_in[0];
    const float* current  = (const float*)d_in[1];
    const float* prev_t   = (const float*)d_in[2];
    const float* latch    = (const float*)d_in[3];
    const float* p_axis   = (const float*)d_in[4];
    const float* tau_t    = (const float*)d_in[5];
    const float* dead_t   = (const float*)d_in[6];
    const float* dbuf     = (const float*)d_in[7];
    const int*   widx     = (const int*)d_in[8];
    float*       out      = (float*)d_out;

    const int n = in_sizes[0];
    const int blocks = (n + TPB * EPT - 1) / (TPB * EPT);
    pam_delay_kernel<<<blocks, TPB, 0, stream>>>(
        target, current, prev_t, latch, p_axis, tau_t, dead_t, dbuf, widx, out, n);
}